// GCNEncoder_35201551958712
// MI455X (gfx1250) — compile-verified
//
#include <hip/hip_runtime.h>
#include <cstdint>

typedef __attribute__((ext_vector_type(2))) float v2f;
typedef __attribute__((ext_vector_type(8))) float v8f;

#define IN_CH  128
#define OUT_CH 64

// ---------------------------------------------------------------------------
// h = x @ W using V_WMMA_F32_16X16X4_F32 (full fp32 matrix path).
// Block = 128 threads = 4 waves; wave w computes N-tile w (cols w*16..w*16+15)
// of one 16-row M-tile. K = 128 -> 32 WMMA ops, fully unrolled.
// ---------------------------------------------------------------------------
__global__ __launch_bounds__(128)
void gcn_gemm_wmma(const float* __restrict__ x, const float* __restrict__ W,
                   float* __restrict__ h, int n_nodes) {
  const int lane  = threadIdx.x & 31;
  const int wave  = threadIdx.x >> 5;     // 0..3 -> N tile
  const int mtile = blockIdx.x;
  const int m16   = lane & 15;
  const int khalf = lane >> 4;            // 0: K pair {0,1}, 1: K pair {2,3}
  int row = (mtile << 4) + m16;
  const int rowc = row < n_nodes ? row : n_nodes - 1;   // tail safety
  const int col  = (wave << 4) + m16;

  const float* xr = x + (size_t)rowc * IN_CH;
  v8f acc = {};
#pragma unroll
  for (int kb = 0; kb < IN_CH; kb += 4) {
    const int k0 = kb + (khalf << 1);
    // A 16x4 f32: lanes 0-15 hold (K=kb,kb+1), lanes 16-31 hold (K=kb+2,kb+3)
    v2f a = *(const v2f*)(xr + k0);
    // B 4x16 f32: lane = N; VGPR0/1 = K pair selected by lane group
    v2f b;
    b.x = W[(size_t)k0 * OUT_CH + col];
    b.y = W[(size_t)(k0 + 1) * OUT_CH + col];
    acc = __builtin_amdgcn_wmma_f32_16x16x4_f32(false, a, false, b,
                                                (short)0, acc, false, false);
  }
  // D 16x16 f32: VGPR v -> M = v + 8*(lane>=16); N = lane&15 within tile
  const int mbase = (mtile << 4) + (khalf << 3);
#pragma unroll
  for (int v = 0; v < 8; ++v) {
    int m = mbase + v;
    if (m < n_nodes) h[(size_t)m * OUT_CH + col] = acc[v];
  }
}

// deg[i] = 1.0f (self loop pre-counted)
__global__ void gcn_deg_init(float* __restrict__ deg, int n) {
  int i = blockIdx.x * blockDim.x + threadIdx.x;
  if (i < n) deg[i] = 1.0f;
}

// out = 0
__global__ void gcn_zero_out(float* __restrict__ out, int total) {
  int i = blockIdx.x * blockDim.x + threadIdx.x;
  if (i < total) out[i] = 0.0f;
}

// deg[col[e]] += 1
__global__ void gcn_deg_count(const int* __restrict__ ei, float* __restrict__ deg,
                              int E) {
  int e = blockIdx.x * blockDim.x + threadIdx.x;
  if (e < E) unsafeAtomicAdd(&deg[ei[E + e]], 1.0f);
}

// deg -> deg^{-1/2}  (deg >= 1 always, self loops)
__global__ void gcn_deg_rsqrt(float* __restrict__ deg, int n) {
  int i = blockIdx.x * blockDim.x + threadIdx.x;
  if (i < n) deg[i] = rsqrtf(deg[i]);
}

// Edge scatter: 16 lanes per edge, float4 gather of h[row], 4 f32 atomics each.
__global__ __launch_bounds__(256)
void gcn_scatter(const float* __restrict__ h, const int* __restrict__ ei,
                 const float* __restrict__ dinv, float* __restrict__ out, int E) {
  long long tid = (long long)blockIdx.x * blockDim.x + threadIdx.x;
  long long total = (long long)E * 16;
  if (tid >= total) return;
  int e = (int)(tid >> 4);
  int c = (int)(tid & 15) << 2;
  int r = ei[e];          // source node
  int t = ei[E + e];      // target node
  float norm = dinv[r] * dinv[t];
  float4 hv = *(const float4*)(h + (size_t)r * OUT_CH + c);
  float* op = out + (size_t)t * OUT_CH + c;
  unsafeAtomicAdd(op + 0, hv.x * norm);
  unsafeAtomicAdd(op + 1, hv.y * norm);
  unsafeAtomicAdd(op + 2, hv.z * norm);
  unsafeAtomicAdd(op + 3, hv.w * norm);
}

// out[i][c] = relu(out[i][c] + h[i][c]*dinv[i]^2 + b[c])   (self loop folded in)
__global__ void gcn_finalize(const float* __restrict__ h,
                             const float* __restrict__ dinv,
                             const float* __restrict__ b,
                             float* __restrict__ out, int n) {
  int i = blockIdx.x * blockDim.x + threadIdx.x;
  if (i >= n * OUT_CH) return;
  int node = i >> 6;            // /64
  int c = i & 63;
  float ds = dinv[node];
  float v = out[i] + h[i] * ds * ds + b[c];
  out[i] = v > 0.0f ? v : 0.0f;
}

extern "C" void kernel_launch(void* const* d_in, const int* in_sizes, int n_in,
                              void* d_out, int out_size, void* d_ws, size_t ws_size,
                              hipStream_t stream) {
  const float* x  = (const float*)d_in[0];
  const int*   ei = (const int*)d_in[1];
  const float* W  = (const float*)d_in[2];
  const float* b  = (const float*)d_in[3];
  const int n = in_sizes[0] / IN_CH;      // 100000
  const int E = in_sizes[1] / 2;          // 1600000
  float* out = (float*)d_out;

  // Workspace layout: h [n*64 f32] | dinv [n f32]
  float* h    = (float*)d_ws;
  float* dinv = h + (size_t)n * OUT_CH;

  const int B = 256;
  // 1) init deg = 1 (self loop), zero out accumulator
  gcn_deg_init<<<(n + B - 1) / B, B, 0, stream>>>(dinv, n);
  gcn_zero_out<<<((size_t)n * OUT_CH + B - 1) / B, B, 0, stream>>>(out, n * OUT_CH);
  // 2) degree counting
  gcn_deg_count<<<(E + B - 1) / B, B, 0, stream>>>(ei, dinv, E);
  // 3) deg^{-1/2}
  gcn_deg_rsqrt<<<(n + B - 1) / B, B, 0, stream>>>(dinv, n);
  // 4) h = x @ W (WMMA)
  gcn_gemm_wmma<<<(n + 15) / 16, 128, 0, stream>>>(x, W, h, n);
  // 5) edge scatter-add
  long long sthreads = (long long)E * 16;
  gcn_scatter<<<(unsigned)((sthreads + B - 1) / B), B, 0, stream>>>(h, ei, dinv, out, E);
  // 6) bias + self-loop + ReLU
  gcn_finalize<<<((size_t)n * OUT_CH + B - 1) / B, B, 0, stream>>>(h, dinv, b, out, n);
}